// PointNetSetAbstraction_90108413870689
// MI455X (gfx1250) — compile-verified
//
#include <hip/hip_runtime.h>
#include <hip/hip_bf16.h>
#include <math.h>

// ---------------------------------------------------------------------------
// PointNet++ GeoConv set-abstraction for MI455X (gfx1250, wave32, WMMA).
//
//  * cosw (400MB) is never materialized. Per (n-tile, m-chunk) it is built
//    ONCE into LDS by all 128 threads (geometry computed once per (n,m)
//    pair, all 6 axis values derived from it), then consumed as WMMA
//    A-fragments via ds_load_b64. This removes the 4x(wave) * 3x(axis)
//    redundant VALU of the naive on-the-fly scheme.
//  * LDS cosw tile padded to stride 68 (== 4 mod 64 banks): a wave's
//    A-fragment ds_load_b64 touches all 64 banks exactly once.
//  * All GEMMs use native fp32 WMMA v_wmma_f32_16x16x4_f32 (reference is
//    fp32; A is generated in fp32 VALU anyway, so low-precision WMMA would
//    add conversions without removing the geometry VALU).
//  * dir_t[b][axis][m][d] (12MB) is fully L2-resident (192MB L2);
//    global_prefetch (builtin_prefetch) pulls the current chunk into the WGP
//    while the LDS fill phase runs.
//  * FPS uses far0=0 (JAX PRNG not reproducible).
// ---------------------------------------------------------------------------

typedef __attribute__((ext_vector_type(2))) float v2f;
typedef __attribute__((ext_vector_type(8))) float v8f;

#define WMMA_F32_16x16x4(a, b, c) \
  __builtin_amdgcn_wmma_f32_16x16x4_f32(false, (a), false, (b), (short)0, (c), false, false)

static constexpr int   B_    = 4;
static constexpr int   N_    = 2048;
static constexpr int   MC_   = 64;    // mid channels (both layers)
static constexpr int   NPT_  = 512;   // npoint
static constexpr int   NS_   = 32;    // nsample
static constexpr float R2_   = 0.01f;           // radius^2
static constexpr float DWDEN = 1.0f / 0.03f;    // 1/((2r)^2 - r^2)

static constexpr int MCH = 64;   // m-chunk staged per iteration
static constexpr int SW  = 68;   // LDS row stride (== 4 mod 64 -> bank-clean)

// ---------------------------------------------------------------------------
// dwsum[b][n] = sum_m max(1 - (|xyz_m - xyz_n|^2 - r^2)/((2r)^2 - r^2), 0)
// ---------------------------------------------------------------------------
__global__ void k_dwsum(const float* __restrict__ xyz, float* __restrict__ dwsum) {
  __shared__ float sx[N_ * 3];
  const int b = blockIdx.y;
  const int n = blockIdx.x * 256 + threadIdx.x;
  const float* X = xyz + (size_t)b * N_ * 3;
  for (int i = threadIdx.x; i < N_ * 3; i += 256) sx[i] = X[i];
  __syncthreads();
  const float px = sx[n * 3 + 0], py = sx[n * 3 + 1], pz = sx[n * 3 + 2];
  float s = 0.0f;
  for (int m = 0; m < N_; ++m) {
    float dx = sx[m * 3 + 0] - px, dy = sx[m * 3 + 1] - py, dz = sx[m * 3 + 2] - pz;
    float v2 = dx * dx + dy * dy + dz * dz;
    s += fmaxf(1.0f - (v2 - R2_) * DWDEN, 0.0f);
  }
  dwsum[b * N_ + n] = s;
}

// ---------------------------------------------------------------------------
// center = points @ cw + cb      (one wave per 16x16 tile, K = lastC)
// ---------------------------------------------------------------------------
__global__ void k_center_gemm(const float* __restrict__ pts, const float* __restrict__ cw,
                              const float* __restrict__ cb, float* __restrict__ center,
                              int lastC, int oc) {
  const int lane = threadIdx.x & 31;
  const int half = lane >> 4, l16 = lane & 15;
  const int col0 = blockIdx.x * 16;
  const int row0 = blockIdx.y * 16;
  const int b    = blockIdx.z;
  const float* A = pts + (size_t)b * N_ * lastC;
  v8f acc = {};
  for (int k = 0; k < lastC; k += 4) {
    v2f af, bf;
    const float* ar = A + (size_t)(row0 + l16) * lastC + k + 2 * half;
    af[0] = ar[0];
    af[1] = ar[1];
    bf[0] = cw[(size_t)(k + 2 * half + 0) * oc + col0 + l16];
    bf[1] = cw[(size_t)(k + 2 * half + 1) * oc + col0 + l16];
    acc = WMMA_F32_16x16x4(af, bf, acc);
  }
  const float bias = cb[col0 + l16];
  float* O = center + ((size_t)b * N_ + row0) * oc + col0 + l16;
#pragma unroll
  for (int i = 0; i < 8; ++i) O[(size_t)(i + 8 * half) * oc] = acc[i] + bias;
}

// ---------------------------------------------------------------------------
// direction = points @ dw + db, scattered to dir_t[b][axis][m][d]
// dw columns are (d*6 + a) per the reference reshape (B,N,mc,6).
// ---------------------------------------------------------------------------
__global__ void k_dir_gemm(const float* __restrict__ pts, const float* __restrict__ dw,
                           const float* __restrict__ db, float* __restrict__ dir_t,
                           int lastC) {
  const int lane = threadIdx.x & 31;
  const int half = lane >> 4, l16 = lane & 15;
  const int col0 = blockIdx.x * 16;     // 0..383 over mc*6
  const int row0 = blockIdx.y * 16;     // m tile
  const int b    = blockIdx.z;
  const float* A = pts + (size_t)b * N_ * lastC;
  v8f acc = {};
  for (int k = 0; k < lastC; k += 4) {
    v2f af, bf;
    const float* ar = A + (size_t)(row0 + l16) * lastC + k + 2 * half;
    af[0] = ar[0];
    af[1] = ar[1];
    bf[0] = dw[(size_t)(k + 2 * half + 0) * (MC_ * 6) + col0 + l16];
    bf[1] = dw[(size_t)(k + 2 * half + 1) * (MC_ * 6) + col0 + l16];
    acc = WMMA_F32_16x16x4(af, bf, acc);
  }
  const int   col  = col0 + l16;
  const int   d    = col / 6;
  const int   a    = col % 6;
  const float bias = db[col];
#pragma unroll
  for (int i = 0; i < 8; ++i) {
    const int m = row0 + i + 8 * half;
    dir_t[(((size_t)b * 6 + a) * N_ + m) * MC_ + d] = acc[i] + bias;
  }
}

// ---------------------------------------------------------------------------
// agg[b][n][d] = sum_{a,m} cosw(b,n,m,a) * dir_t[b][a][m][d]
//
// Block = 128 threads (4 waves). Per m-chunk of 64:
//   phase 1 (all threads): build scw[a][nl][ml] in LDS -- geometry computed
//            once per (n,m) pair, 6 axis values derived.
//   phase 2 (per wave, d-slab = wid*16): WMMA K-loop, A from LDS
//            (ds_load_b64, bank-conflict-free via SW=68), B from global
//            (L2-resident dir_t, prefetched during phase 1).
// ---------------------------------------------------------------------------
__global__ void k_agg(const float* __restrict__ xyz, const float* __restrict__ dwsum,
                      const float* __restrict__ dir_t, float* __restrict__ agg) {
  __shared__ float sx[N_ * 3];        // 24KB: xyz[b]
  __shared__ float scw[6 * 16 * SW];  // ~25.5KB: cosw tile
  __shared__ float srw[16];           // 1/dwsum for the 16 n-rows
  const int lane = threadIdx.x & 31;
  const int wid  = threadIdx.x >> 5;  // 0..3 -> d-slab
  const int half = lane >> 4, l16 = lane & 15;
  const int d0 = wid * 16;
  const int n0 = blockIdx.y * 16;
  const int b  = blockIdx.z;
  const float* X = xyz + (size_t)b * N_ * 3;
  for (int i = threadIdx.x; i < N_ * 3; i += blockDim.x) sx[i] = X[i];
  if (threadIdx.x < 16) srw[threadIdx.x] = 1.0f / dwsum[b * N_ + n0 + threadIdx.x];
  __syncthreads();

  v8f acc = {};
  for (int mb = 0; mb < N_; mb += MCH) {
    // --- prefetch the 6 dir_t axis slabs for this chunk toward the WGP ---
#pragma unroll
    for (int a = 0; a < 6; ++a) {
      const float* base = dir_t + (((size_t)b * 6 + a) * N_ + mb) * MC_;
      for (int q = threadIdx.x * 32; q < MCH * MC_; q += 128 * 32)
        __builtin_prefetch(base + q, 0, 3);
    }
    // --- phase 1: cooperative cosw fill (geometry once per pair) ---
    for (int idx = threadIdx.x; idx < 16 * MCH; idx += 128) {
      const int nl = idx / MCH;
      const int ml = idx % MCH;
      const int nn = n0 + nl;
      const int m  = mb + ml;
      const float dx = sx[m * 3 + 0] - sx[nn * 3 + 0];
      const float dy = sx[m * 3 + 1] - sx[nn * 3 + 1];
      const float dz = sx[m * 3 + 2] - sx[nn * 3 + 2];
      const float v2  = dx * dx + dy * dy + dz * dz;
      const float vn  = sqrtf(v2);
      const float inv = 1.0f / (vn + 1e-8f);
      const float wgt = fmaxf(1.0f - (v2 - R2_) * DWDEN, 0.0f) * srw[nl];
      const float uz = dz * inv, uy = dy * inv, ux = dx * inv;
      float t;
      t = fmaxf(uz, 0.0f);  scw[(0 * 16 + nl) * SW + ml] = t * t * wgt;
      t = fmaxf(-uz, 0.0f); scw[(1 * 16 + nl) * SW + ml] = t * t * wgt;
      t = fmaxf(uy, 0.0f);  scw[(2 * 16 + nl) * SW + ml] = t * t * wgt;
      t = fmaxf(-uy, 0.0f); scw[(3 * 16 + nl) * SW + ml] = t * t * wgt;
      t = fmaxf(ux, 0.0f);  scw[(4 * 16 + nl) * SW + ml] = t * t * wgt;
      t = fmaxf(-ux, 0.0f); scw[(5 * 16 + nl) * SW + ml] = t * t * wgt;
    }
    __syncthreads();
    // --- phase 2: WMMA over the staged chunk ---
    for (int a = 0; a < 6; ++a) {
      const float* Arow = scw + (a * 16 + l16) * SW;
      const float* Bcol = dir_t + (((size_t)b * 6 + a) * N_ + mb) * MC_ + d0 + l16;
#pragma unroll 4
      for (int m0 = 0; m0 < MCH; m0 += 4) {
        const int mo = m0 + 2 * half;
        v2f af, bf;
        af[0] = Arow[mo];
        af[1] = Arow[mo + 1];
        bf[0] = Bcol[(size_t)(mo + 0) * MC_];
        bf[1] = Bcol[(size_t)(mo + 1) * MC_];
        acc = WMMA_F32_16x16x4(af, bf, acc);
      }
    }
    __syncthreads();
  }
  float* O = agg + ((size_t)b * N_ + n0) * MC_ + d0 + l16;
#pragma unroll
  for (int i = 0; i < 8; ++i) O[(size_t)(i + 8 * half) * MC_] = acc[i];
}

// ---------------------------------------------------------------------------
// Batch-norm statistics over (B,N) per channel; population variance.
// ---------------------------------------------------------------------------
__global__ void k_bn_stats(const float* __restrict__ x, int C,
                           float* __restrict__ mean, float* __restrict__ var) {
  __shared__ float ssum[256], ssq[256];
  const int c   = blockIdx.x;
  const int tid = threadIdx.x;
  float s = 0.0f, q = 0.0f;
  for (int i = tid; i < B_ * N_; i += 256) {
    const float v = x[(size_t)i * C + c];
    s += v;
    q += v * v;
  }
  ssum[tid] = s; ssq[tid] = q;
  __syncthreads();
  for (int st = 128; st > 0; st >>= 1) {
    if (tid < st) { ssum[tid] += ssum[tid + st]; ssq[tid] += ssq[tid + st]; }
    __syncthreads();
  }
  if (tid == 0) {
    const float m = ssum[0] * (1.0f / (B_ * N_));
    mean[c] = m;
    var[c]  = ssq[0] * (1.0f / (B_ * N_)) - m * m;
  }
}

__global__ void k_bn_relu(const float* __restrict__ x, const float* __restrict__ mean,
                          const float* __restrict__ var, const float* __restrict__ g,
                          const float* __restrict__ bt, float* __restrict__ y,
                          int C, int total) {
  const int i = blockIdx.x * 256 + threadIdx.x;
  if (i >= total) return;
  const int c = i % C;
  const float v = (x[i] - mean[c]) * rsqrtf(var[c] + 1e-5f) * g[c] + bt[c];
  y[i] = fmaxf(v, 0.0f);
}

// ---------------------------------------------------------------------------
// pre = center + (agg @ d2w + d2b)     (K = MC_ = 64)
// ---------------------------------------------------------------------------
__global__ void k_d2_center(const float* __restrict__ agg, const float* __restrict__ d2w,
                            const float* __restrict__ d2b, const float* __restrict__ center,
                            float* __restrict__ pre, int oc) {
  const int lane = threadIdx.x & 31;
  const int half = lane >> 4, l16 = lane & 15;
  const int col0 = blockIdx.x * 16;
  const int row0 = blockIdx.y * 16;
  const int b    = blockIdx.z;
  const float* A = agg + (size_t)b * N_ * MC_;
  v8f acc = {};
#pragma unroll 4
  for (int k = 0; k < MC_; k += 4) {
    v2f af, bf;
    const float* ar = A + (size_t)(row0 + l16) * MC_ + k + 2 * half;
    af[0] = ar[0];
    af[1] = ar[1];
    bf[0] = d2w[(size_t)(k + 2 * half + 0) * oc + col0 + l16];
    bf[1] = d2w[(size_t)(k + 2 * half + 1) * oc + col0 + l16];
    acc = WMMA_F32_16x16x4(af, bf, acc);
  }
  const float bias = d2b[col0 + l16];
  const size_t base = ((size_t)b * N_ + row0) * oc + col0 + l16;
#pragma unroll
  for (int i = 0; i < 8; ++i) {
    const size_t o = base + (size_t)(i + 8 * half) * oc;
    pre[o] = acc[i] + bias + center[o];
  }
}

// ---------------------------------------------------------------------------
// Farthest point sampling (sequential 512-step scan; far0 = 0 since the
// JAX PRNG seed is not reproducible here). Writes fps indices + new_xyz.
// ---------------------------------------------------------------------------
__global__ void k_fps(const float* __restrict__ xyz, int* __restrict__ fps_idx,
                      float* __restrict__ newxyz) {
  __shared__ float sx[N_ * 3];
  __shared__ float dist[N_];
  __shared__ float rmax[256];
  __shared__ int   rarg[256];
  const int b = blockIdx.x, tid = threadIdx.x;
  const float* X = xyz + (size_t)b * N_ * 3;
  for (int i = tid; i < N_ * 3; i += 256) sx[i] = X[i];
  for (int i = tid; i < N_; i += 256) dist[i] = 1e10f;
  __syncthreads();
  int far = 0;
  for (int it = 0; it < NPT_; ++it) {
    if (tid == 0) {
      fps_idx[b * NPT_ + it] = far;
      newxyz[((size_t)b * NPT_ + it) * 3 + 0] = sx[far * 3 + 0];
      newxyz[((size_t)b * NPT_ + it) * 3 + 1] = sx[far * 3 + 1];
      newxyz[((size_t)b * NPT_ + it) * 3 + 2] = sx[far * 3 + 2];
    }
    const float cx = sx[far * 3 + 0], cy = sx[far * 3 + 1], cz = sx[far * 3 + 2];
    float bm = -1.0f; int ba = 0;
    for (int i = tid; i < N_; i += 256) {
      const float dx = sx[i * 3 + 0] - cx, dy = sx[i * 3 + 1] - cy, dz = sx[i * 3 + 2] - cz;
      const float d  = dx * dx + dy * dy + dz * dz;
      const float nd = fminf(dist[i], d);
      dist[i] = nd;
      if (nd > bm) { bm = nd; ba = i; }
    }
    rmax[tid] = bm; rarg[tid] = ba;
    __syncthreads();
    for (int st = 128; st > 0; st >>= 1) {
      if (tid < st && rmax[tid + st] > rmax[tid]) { rmax[tid] = rmax[tid + st]; rarg[tid] = rarg[tid + st]; }
      __syncthreads();
    }
    far = rarg[0];
    __syncthreads();
  }
}

// ---------------------------------------------------------------------------
// Ball query: first NS_ indices (ascending) within radius, padded with first.
// ---------------------------------------------------------------------------
__global__ void k_ball(const float* __restrict__ xyz, const float* __restrict__ newxyz,
                       int* __restrict__ ball_idx) {
  const int t = blockIdx.x * 256 + threadIdx.x;
  if (t >= B_ * NPT_) return;
  const int b = t / NPT_;
  const float* X = xyz + (size_t)b * N_ * 3;
  const float qx = newxyz[t * 3 + 0], qy = newxyz[t * 3 + 1], qz = newxyz[t * 3 + 2];
  int* out = ball_idx + (size_t)t * NS_;
  int cnt = 0, first = 0;
  for (int j = 0; j < N_ && cnt < NS_; ++j) {
    const float dx = X[j * 3 + 0] - qx, dy = X[j * 3 + 1] - qy, dz = X[j * 3 + 2] - qz;
    if (dx * dx + dy * dy + dz * dz <= R2_) {
      if (cnt == 0) first = j;
      out[cnt++] = j;
    }
  }
  for (; cnt < NS_; ++cnt) out[cnt] = first;
}

// ---------------------------------------------------------------------------
// Gather grouped points and max-pool over nsample.
// ---------------------------------------------------------------------------
__global__ void k_groupmax(const float* __restrict__ pts, const int* __restrict__ ball_idx,
                           float* __restrict__ outp, int C) {
  const int t = blockIdx.x * 256 + threadIdx.x;
  if (t >= B_ * NPT_ * C) return;
  const int c = t % C;
  const int s = (t / C) % NPT_;
  const int b = t / (C * NPT_);
  const int* idx = ball_idx + ((size_t)b * NPT_ + s) * NS_;
  const float* P = pts + (size_t)b * N_ * C;
  float m = -3.4028235e38f;
#pragma unroll 4
  for (int k = 0; k < NS_; ++k) m = fmaxf(m, P[(size_t)idx[k] * C + c]);
  outp[t] = m;
}

// ---------------------------------------------------------------------------
// Host orchestration
// ---------------------------------------------------------------------------
static void run_geoconv(const float* xyz, const float* dwsum, const float* inpts,
                        int lastC, int oc,
                        const float* cw, const float* cb, const float* dw, const float* db,
                        const float* d2w, const float* d2b,
                        const float* g1, const float* b1, const float* g2, const float* b2,
                        float* dirT, float* center, float* agg, float* pre, float* outpts,
                        float* mean, float* var, hipStream_t stream) {
  k_center_gemm<<<dim3(oc / 16, N_ / 16, B_), 32, 0, stream>>>(inpts, cw, cb, center, lastC, oc);
  k_dir_gemm<<<dim3((MC_ * 6) / 16, N_ / 16, B_), 32, 0, stream>>>(inpts, dw, db, dirT, lastC);
  k_agg<<<dim3(1, N_ / 16, B_), 128, 0, stream>>>(xyz, dwsum, dirT, agg);
  k_bn_stats<<<MC_, 256, 0, stream>>>(agg, MC_, mean, var);
  {
    const int tot = B_ * N_ * MC_;
    k_bn_relu<<<(tot + 255) / 256, 256, 0, stream>>>(agg, mean, var, g1, b1, agg, MC_, tot);
  }
  k_d2_center<<<dim3(oc / 16, N_ / 16, B_), 32, 0, stream>>>(agg, d2w, d2b, center, pre, oc);
  k_bn_stats<<<oc, 256, 0, stream>>>(pre, oc, mean, var);
  {
    const int tot = B_ * N_ * oc;
    k_bn_relu<<<(tot + 255) / 256, 256, 0, stream>>>(pre, mean, var, g2, b2, outpts, oc, tot);
  }
}

extern "C" void kernel_launch(void* const* d_in, const int* in_sizes, int n_in,
                              void* d_out, int out_size, void* d_ws, size_t ws_size,
                              hipStream_t stream) {
  (void)in_sizes; (void)n_in; (void)out_size; (void)ws_size;

  const float* xyz  = (const float*)d_in[0];
  const float* pts0 = (const float*)d_in[1];
  // params layer0: d_in[2..11], layer1: d_in[12..21]
  // order per dict insertion: cw, cb, dw, db, d2w, d2b, g1, b1, g2, b2

  float* W      = (float*)d_ws;
  float* dwsum  = W;                       // 4*2048                  =     8192
  float* dirT   = dwsum + 8192;            // 4*6*2048*64             =  3145728
  float* center = dirT + 3145728;          // 4*2048*128 (max oc)     =  1048576
  float* agg    = center + 1048576;        // 4*2048*64               =   524288
  float* pre    = agg + 524288;            // 4*2048*128              =  1048576
  float* ptsA   = pre + 1048576;           // 4*2048*64  (layer0 out) =   524288
  float* ptsB   = ptsA + 524288;           // 4*2048*128 (layer1 out) =  1048576
  float* mean   = ptsB + 1048576;          // 128
  float* var    = mean + 128;              // 128
  int*   fpsI   = (int*)(var + 128);       // 4*512
  int*   ballI  = fpsI + B_ * NPT_;        // 4*512*32

  float* out_newxyz = (float*)d_out;                   // 4*512*3
  float* out_newpts = out_newxyz + B_ * NPT_ * 3;      // 4*512*128

  // Shared decay-weight normalizer (cosw row sums), reused by both layers.
  k_dwsum<<<dim3(N_ / 256, B_), 256, 0, stream>>>(xyz, dwsum);

  // Layer 0: in=32ch, mid=64, out=64
  {
    const float** p = (const float**)(d_in + 2);
    run_geoconv(xyz, dwsum, pts0, 32, 64,
                p[0], p[1], p[2], p[3], p[4], p[5], p[6], p[7], p[8], p[9],
                dirT, center, agg, pre, ptsA, mean, var, stream);
  }
  // Layer 1: in=64ch, mid=64, out=128
  {
    const float** p = (const float**)(d_in + 12);
    run_geoconv(xyz, dwsum, ptsA, 64, 128,
                p[0], p[1], p[2], p[3], p[4], p[5], p[6], p[7], p[8], p[9],
                dirT, center, agg, pre, ptsB, mean, var, stream);
  }

  // Sampling + grouping + max pool
  k_fps<<<B_, 256, 0, stream>>>(xyz, fpsI, out_newxyz);
  k_ball<<<(B_ * NPT_ + 255) / 256, 256, 0, stream>>>(xyz, out_newxyz, ballI);
  {
    const int tot = B_ * NPT_ * 128;
    k_groupmax<<<(tot + 255) / 256, 256, 0, stream>>>(ptsB, ballI, out_newpts, 128);
  }
}